// HPSS_35510789604138
// MI455X (gfx1250) — compile-verified
//
#include <hip/hip_runtime.h>
#include <hip/hip_bf16.h>
#include <stdint.h>

// ---------------- problem constants (from reference) ----------------
#define FDIM 1025
#define TDIM 1024
#define NIMG 4            // B*C = 2*2
#define OFF  15           // KERNEL//2, KERNEL = 31
#define EPSV 1e-6f

// ---------------- tiling ----------------
#define TF 32             // output rows (F) per workgroup
#define TT 64             // output cols (T) per workgroup
#define PA TT             // sA pitch (floats)
#define PB (TT + 2*OFF)   // sB pitch = 94
#define AROWS (TF + 2*OFF)// 62

__device__ __forceinline__ int reflect_idx(int i, int n) {
    i = (i < 0) ? -i : i;
    i = (i >= n) ? (2*n - 2 - i) : i;
    return i;
}

__device__ __forceinline__ void cswap(float& a, float& b) {
    float t = fminf(a, b);
    b = fmaxf(a, b);
    a = t;
}

// Select-min-and-max network over v[0..S-1]:
// places min at v[0], max at v[S-1], preserves the multiset of the rest.
// Cost ~3(S-1)/2 cswaps (vs 2S-3 for bubble passes).
template<int S>
__device__ __forceinline__ void minmax_head_tail(float* v) {
    // stage 1: pairwise — pair-min in even slot, pair-max in odd slot
#pragma unroll
    for (int i = 0; i + 1 < S; i += 2) cswap(v[i], v[i + 1]);
    if constexpr ((S & 1) == 0) {
        // min chain over even slots -> v[0]
#pragma unroll
        for (int i = 2; i < S; i += 2) cswap(v[0], v[i]);
        // max chain over odd slots -> v[S-1]
#pragma unroll
        for (int i = 1; i + 2 < S; i += 2) cswap(v[i], v[i + 2]);
    } else {
        // pairs cover v[0..S-2]; v[S-1] is unpaired (could be min or max)
#pragma unroll
        for (int i = 2; i < S - 1; i += 2) cswap(v[0], v[i]);
        cswap(v[0], v[S - 1]);              // fold unpaired into min; its max side stays
#pragma unroll
        for (int i = 1; i + 2 < S - 1; i += 2) cswap(v[i], v[i + 2]); // max -> v[S-2]
        cswap(v[S - 2], v[S - 1]);          // global max -> v[S-1]
    }
}

// One forgetful-selection round at active size S, then recurse to S-1.
// Round: extract min (v[0]) & max (v[S-1]), discard both, refill v[0] with
// the next window element. Stops when 3 survivors remain.
template<int S>
struct Forget {
    static __device__ __forceinline__ void run(float* v, const float* __restrict__ p,
                                               int stride, int e) {
        minmax_head_tail<S>(v);
        v[0] = p[e * stride];
        Forget<S - 1>::run(v, p, stride, e + 1);
    }
};
template<>
struct Forget<3> {
    static __device__ __forceinline__ void run(float*, const float* __restrict__, int, int) {}
};

// Median of 31 values p[0], p[stride], ..., p[30*stride].
// Forgetful selection with buffer ceil(31/2)+1 = 17.
__device__ __forceinline__ float median31(const float* __restrict__ p, int stride) {
    float v[17];
#pragma unroll
    for (int j = 0; j < 17; ++j) v[j] = p[j * stride];
    Forget<17>::run(v, p, stride, 17);      // 14 rounds, consumes elements 17..30
    return __builtin_amdgcn_fmed3f(v[0], v[1], v[2]);   // v_med3_f32
}

// ---------------- CDNA5 async global -> LDS copies (ASYNCcnt) ----------------
__device__ __forceinline__ void async_g2lds_b32(uint32_t lds_off, const float* gptr) {
    uint64_t ga = (uint64_t)(uintptr_t)gptr;
    asm volatile("global_load_async_to_lds_b32 %0, %1, off"
                 :: "v"(lds_off), "v"(ga) : "memory");
}
__device__ __forceinline__ void async_g2lds_b128(uint32_t lds_off, const float* gptr) {
    uint64_t ga = (uint64_t)(uintptr_t)gptr;
    asm volatile("global_load_async_to_lds_b128 %0, %1, off"
                 :: "v"(lds_off), "v"(ga) : "memory");
}
__device__ __forceinline__ void wait_async0() {
    asm volatile("s_wait_asynccnt 0x0" ::: "memory");
}

__global__ void __launch_bounds__(256)
hpss_median_kernel(const float* __restrict__ in, float* __restrict__ out)
{
    __shared__ float sA[AROWS * PA];   // F-extended tile (62 x 64), reflect rows
    __shared__ float sB[TF * PB];      // T-extended tile (32 x 94), reflect cols

    const int t0  = blockIdx.x * TT;
    const int f0  = blockIdx.y * TF;
    const int img = blockIdx.z;
    const size_t imgOff = (size_t)img * FDIM * TDIM;
    const float* base = in + imgOff;
    const int tid = threadIdx.x;

    // ---- fill sA: rows f0-15 .. f0+46 (reflected), cols t0..t0+63 (in range) ----
    // 62 rows x 16 vec4 = 992 b128 async copies; row base is 16B aligned (T=1024).
    for (int idx = tid; idx < AROWS * 16; idx += 256) {
        const int row = idx >> 4;
        const int c4  = (idx & 15) << 2;
        const int gr  = reflect_idx(f0 - OFF + row, FDIM);
        const uint32_t loff = (uint32_t)(uintptr_t)&sA[row * PA + c4];
        async_g2lds_b128(loff, base + (size_t)gr * TDIM + (t0 + c4));
    }
    // ---- fill sB: rows f0..f0+31 (clamped; invalid rows unused), cols reflected ----
    for (int idx = tid; idx < TF * PB; idx += 256) {
        const int row = idx / PB;
        const int col = idx - row * PB;
        int grow = f0 + row; if (grow > FDIM - 1) grow = FDIM - 1;
        const int gcol = reflect_idx(t0 - OFF + col, TDIM);
        const uint32_t loff = (uint32_t)(uintptr_t)&sB[idx];
        async_g2lds_b32(loff, base + (size_t)grow * TDIM + gcol);
    }
    wait_async0();        // my wave's async LDS writes landed
    __syncthreads();      // everyone's landed

    const int tl  = tid & 63;   // col within tile
    const int flb = tid >> 6;   // 0..3
    const size_t S = (size_t)NIMG * FDIM * TDIM;   // one output plane

#pragma unroll 1
    for (int it = 0; it < 8; ++it) {
        const int fl = flb + (it << 2);
        const int f  = f0 + fl;
        if (f >= FDIM) continue;

        // perc: median along F (stride = PA), harm: median along T (stride = 1)
        const float pm = median31(&sA[fl * PA + tl], PA);
        const float hm = median31(&sB[fl * PB + tl], 1);
        const float p  = pm * pm;           // POWER = 2.0
        const float h  = hm * hm;
        const float x  = sA[(fl + OFF) * PA + tl];   // center magnitude

        const float rd = 1.0f / (h + p + EPSV);
        const float mh = (h + EPSV) * rd;
        const float mp = (p + EPSV) * rd;

        const size_t o = imgOff + (size_t)f * TDIM + (t0 + tl);
        __builtin_nontemporal_store(x * mh, &out[o]);          // mag * mask_harm
        __builtin_nontemporal_store(x * mp, &out[o + S]);      // mag * mask_perc
        __builtin_nontemporal_store(mh,     &out[o + 2 * S]);  // mask_harm
        __builtin_nontemporal_store(mp,     &out[o + 3 * S]);  // mask_perc
    }
}

extern "C" void kernel_launch(void* const* d_in, const int* in_sizes, int n_in,
                              void* d_out, int out_size, void* d_ws, size_t ws_size,
                              hipStream_t stream) {
    (void)in_sizes; (void)n_in; (void)out_size; (void)d_ws; (void)ws_size;
    const float* in = (const float*)d_in[0];
    float* out = (float*)d_out;
    dim3 grid(TDIM / TT, (FDIM + TF - 1) / TF, NIMG);   // (16, 33, 4)
    hpss_median_kernel<<<grid, 256, 0, stream>>>(in, out);
}